// Pool_12532714569792
// MI455X (gfx1250) — compile-verified
//
#include <hip/hip_runtime.h>

// Cumulative mean along seq axis: out[b,s,d] = cumsum(x,axis=1)/(s+1)
// x: f32[8, 4096, 1024]. Memory-bound (256 MiB traffic, ~11us @ 23.3 TB/s).
// Two-phase block scan; cross-segment exclusive scan done with
// V_WMMA_F32_16X16X4_F32 (D = strict-lower-triangular-ones x segment-sums).

typedef float v4f __attribute__((ext_vector_type(4)));
typedef float v2f __attribute__((ext_vector_type(2)));
typedef float v8f __attribute__((ext_vector_type(8)));

constexpr int S   = 4096;      // seq length
constexpr int Dm  = 1024;      // model dim
constexpr int TX  = 16;        // column-group threads (4 cols each => 64 cols/block)
constexpr int TY  = 16;        // segments along seq
constexpr int SEG = S / TY;    // 256 rows per segment
constexpr int DT  = TX * 4;    // 64 columns per block

__global__ __launch_bounds__(TX * TY) void cummean_kernel(
    const float* __restrict__ x, float* __restrict__ out) {
  __shared__ float lds_sums[DT][TY + 1];  // padded: avoid bank conflicts
  __shared__ float lds_excl[DT][TY + 1];
  __shared__ float inv_lds[S];            // 1/(s+1) table

  const int tx  = threadIdx.x;
  const int ty  = threadIdx.y;
  const int tid = ty * TX + tx;

  // Cooperative fill of reciprocal table (16 IEEE divides per thread, once).
  for (int i = tid; i < S; i += TX * TY) {
    inv_lds[i] = 1.0f / (float)(i + 1);
  }

  const int tile        = blockIdx.x;       // 0..127
  const int tiles_per_b = Dm / DT;          // 16 (tiles never straddle a batch)
  const int b           = tile / tiles_per_b;
  const int dbase       = (tile % tiles_per_b) * DT;
  const float* xb = x   + (size_t)b * S * Dm + dbase;
  float*       ob = out + (size_t)b * S * Dm + dbase;

  const int dd = tx * 4;  // 4 columns per thread via float4

  // ---- Pass 1: per-segment sums (streaming, coalesced float4) ----
  v4f acc = {0.f, 0.f, 0.f, 0.f};
  {
    const v4f* p = (const v4f*)(xb + (size_t)(ty * SEG) * Dm + dd);
    #pragma unroll 4
    for (int i = 0; i < SEG; ++i) {
      acc += p[(size_t)i * (Dm / 4)];
    }
  }
  lds_sums[dd + 0][ty] = acc.x;
  lds_sums[dd + 1][ty] = acc.y;
  lds_sums[dd + 2][ty] = acc.z;
  lds_sums[dd + 3][ty] = acc.w;
  __syncthreads();

  // ---- Cross-segment EXCLUSIVE scan via WMMA: D = L' (16x16, strict lower
  // triangular ones) x X (16 segs x 16 cols). K=16 done as 4 chained
  // v_wmma_f32_16x16x4_f32 accumulating through C. One 16-col group per wave.
  const int wid  = tid >> 5;   // wave id (branch is wave-uniform, EXEC all-1s)
  const int lane = tid & 31;
  if (wid < DT / 16) {         // waves 0..3 cover 64 columns
    const int colbase = wid * 16;
    const int m    = lane & 15;   // A row / B column index
    const int half = lane >> 4;
    v8f c = {};
    #pragma unroll
    for (int s4 = 0; s4 < 4; ++s4) {
      // A 16x4 f32 layout: VGPR j holds K = s4*4 + 2*half + j, row m.
      const int k0 = s4 * 4 + half * 2;
      v2f a, bm;
      a.x = (k0     < m) ? 1.0f : 0.0f;   // strict lower triangle -> exclusive
      a.y = (k0 + 1 < m) ? 1.0f : 0.0f;
      // B 4x16 f32: VGPR j holds row K = s4*4 + 2*half + j, col = lane%16.
      bm.x = lds_sums[colbase + m][k0];
      bm.y = lds_sums[colbase + m][k0 + 1];
      c = __builtin_amdgcn_wmma_f32_16x16x4_f32(false, a, false, bm, (short)0,
                                                c, false, false);
    }
    // D 16x16 f32: VGPR r holds seg = r + 8*half, col = lane%16.
    #pragma unroll
    for (int r = 0; r < 8; ++r) {
      lds_excl[colbase + m][r + 8 * half] = c[r];
    }
  }
  __syncthreads();

  // ---- Pass 2: in-segment running prefix + WMMA offset, scale by 1/(s+1).
  // Re-read streams from L2 (input fits in 192MB L2); NT stores keep the
  // output stream from evicting the L2-resident input.
  v4f run;
  run.x = lds_excl[dd + 0][ty];
  run.y = lds_excl[dd + 1][ty];
  run.z = lds_excl[dd + 2][ty];
  run.w = lds_excl[dd + 3][ty];

  const int s0 = ty * SEG;
  const v4f* p = (const v4f*)(xb + (size_t)s0 * Dm + dd);
  v4f*       q = (v4f*)(ob + (size_t)s0 * Dm + dd);
  #pragma unroll 4
  for (int i = 0; i < SEG; ++i) {
    v4f v = __builtin_nontemporal_load(&p[(size_t)i * (Dm / 4)]);
    run += v;
    const float inv = inv_lds[s0 + i];
    v4f o = run * inv;
    __builtin_nontemporal_store(o, &q[(size_t)i * (Dm / 4)]);
  }
}

extern "C" void kernel_launch(void* const* d_in, const int* in_sizes, int n_in,
                              void* d_out, int out_size, void* d_ws, size_t ws_size,
                              hipStream_t stream) {
  const float* x = (const float*)d_in[0];
  float* out = (float*)d_out;
  const int B = in_sizes[0] / (S * Dm);      // 8
  dim3 block(TX, TY);                         // 256 threads = 8 waves
  dim3 grid((unsigned)(B * Dm / DT));         // 128 blocks
  hipLaunchKernelGGL(cummean_kernel, grid, block, 0, stream, x, out);
}